// Net_12137577579204
// MI455X (gfx1250) — compile-verified
//
#include <hip/hip_runtime.h>
#include <hip/hip_bf16.h>

typedef float v2f __attribute__((ext_vector_type(2)));
typedef float v8f __attribute__((ext_vector_type(8)));

// Problem constants (from reference)
constexpr int kH     = 256;    // hidden
constexpr int kOUT   = 128;
constexpr int kLV    = 4;
constexpr int kT     = 2048;   // sequences per level
constexpr int kL     = 32;     // max seq length
constexpr int kN     = 262144;
constexpr int kNB    = 65536;
constexpr int kB     = 64;
constexpr int kGates = 4 * kH; // 1024
constexpr int kKtot  = 2 * kH; // 512  ( [x_t | h] )

// GEMM tiling: 8 waves, each computing a 32x32 tile (2x2 register blocking)
constexpr int BM = 64;    // block rows (sequences)
constexpr int BN = 128;   // block cols (gate outputs)
constexpr int KC = 64;    // K chunk staged in LDS

// ---------------------------------------------------------------------------
// utility kernels
// ---------------------------------------------------------------------------
__global__ void zero_f(float* __restrict__ p, int n) {
  int g = blockIdx.x * blockDim.x + threadIdx.x;
  if (g < n) p[g] = 0.0f;
}

__global__ void init_bags(const float* __restrict__ bag_emb,
                          const int* __restrict__ x_bag,
                          float* __restrict__ xattr,
                          float* __restrict__ xposb) {
  int g = blockIdx.x * blockDim.x + threadIdx.x;   // NB*H threads
  int b  = g >> 8;
  int hh = g & 255;
  xattr[g] = bag_emb[x_bag[b] * kH + hh];
  xposb[g] = 0.0f;
}

// x_node = emb0[x0] + emb1[x1]; pos_feat = x_pos @ pos_W^T + pos_b
// scatter-add both into bag matrices keyed by col[n]
__global__ __launch_bounds__(256)
void node_scatter(const float* __restrict__ emb0,
                  const float* __restrict__ emb1,
                  const float* __restrict__ posW,   // [H,16]
                  const float* __restrict__ posb,   // [H]
                  const float* __restrict__ xpos,   // [N,16]
                  const int*   __restrict__ xcat,   // [N,2]
                  const int*   __restrict__ col,    // [N]
                  float* __restrict__ xattr,
                  float* __restrict__ xposb) {
  unsigned g = blockIdx.x * 256u + threadIdx.x;   // N*H threads (67M < 2^31)
  int n  = g >> 8;
  int hh = g & 255;
  int c  = col[n];
  float xn = emb0[xcat[2 * n + 0] * kH + hh] + emb1[xcat[2 * n + 1] * kH + hh];
  atomicAdd(&xattr[(size_t)c * kH + hh], xn);
  float pf = posb[hh];
  const float* xp = xpos + (size_t)n * 16;
  const float* pw = posW + (size_t)hh * 16;
#pragma unroll
  for (int j = 0; j < 16; ++j) pf += xp[j] * pw[j];
  atomicAdd(&xposb[(size_t)c * kH + hh], pf);
}

// ---------------------------------------------------------------------------
// LSTM gate GEMM: gates[m, 0:1024] = [x_t(m) | h(m)] @ [Wih;Whh]^T + bih + bhh
// x_t(m) = src[idx[m*32+t], :]   (gather from bag matrix)
// fp32 WMMA 16x16x4; each wave owns a 32x32 tile (4 accumulators).
// A and B both stored K-contiguous in LDS so every fragment is one
// ds_load_2addr_b32 (offset1:1) -> pure 4x ds_load + 4x wmma steady state.
// ---------------------------------------------------------------------------
__global__ __launch_bounds__(256)
void lstm_gates_gemm(const float* __restrict__ src,   // [NB, H]
                     const int*   __restrict__ idx,   // [T, 32]
                     const float* __restrict__ h,     // [T, H]
                     const float* __restrict__ Wih,   // [4H, H]
                     const float* __restrict__ Whh,   // [4H, H]
                     const float* __restrict__ bih,   // [4H]
                     const float* __restrict__ bhh,   // [4H]
                     float* __restrict__ gates,       // [T, 4H]
                     int t) {
  __shared__ float As[BM][KC + 1];   // row m, K-contiguous; +1 pad
  __shared__ float Bs[BN][KC + 1];   // col n, K-contiguous; +1 pad
  __shared__ int   rowbag[BM];

  const int tid = threadIdx.x;
  const int bm0 = blockIdx.x * BM;
  const int bn0 = blockIdx.y * BN;

  if (tid < BM) rowbag[tid] = idx[(bm0 + tid) * kL + t];
  __syncthreads();

  const int wave = tid >> 5;
  const int lane = tid & 31;
  const int half = lane >> 4;   // 0: K pair {0,1} / M rows 0-7 ; 1: {2,3} / rows 8-15
  const int lr   = lane & 15;
  const int wm   = wave >> 2;   // 0..1  -> 32-row band
  const int wn   = wave & 3;    // 0..3  -> 32-col band

  const int ar0 = wm * 32 + lr;        // A row for row-tile 0
  const int ar1 = ar0 + 16;            // A row for row-tile 1
  const int bc0 = wn * 32 + lr;        // B col for col-tile 0
  const int bc1 = bc0 + 16;            // B col for col-tile 1

  v8f acc00 = {}, acc01 = {}, acc10 = {}, acc11 = {};

#pragma unroll 1
  for (int kc = 0; kc < kKtot; kc += KC) {
    // stage A: 64 rows x 64 k (coalesced along k, contiguous LDS writes)
#pragma unroll
    for (int j = 0; j < (BM * KC) / 256; ++j) {
      int e  = tid + j * 256;
      int r  = e >> 6;
      int kk = e & 63;
      int k  = kc + kk;
      float v;
      if (k < kH) v = src[(size_t)rowbag[r] * kH + k];
      else        v = h[(size_t)(bm0 + r) * kH + (k - kH)];
      As[r][kk] = v;
    }
    // stage B: 128 cols x 64 k (coalesced along k, contiguous LDS writes)
#pragma unroll
    for (int j = 0; j < (BN * KC) / 256; ++j) {
      int e  = tid + j * 256;
      int nn = e >> 6;
      int kk = e & 63;
      int k  = kc + kk;
      int n  = bn0 + nn;
      float v;
      if (k < kH) v = Wih[(size_t)n * kH + k];
      else        v = Whh[(size_t)n * kH + (k - kH)];
      Bs[nn][kk] = v;
    }
    __syncthreads();

#pragma unroll
    for (int k0 = 0; k0 < KC; k0 += 4) {
      v2f a0, a1, b0, b1;
      // A 16x4 f32 layout: lanes 0-15 hold K={0,1}, lanes 16-31 K={2,3}; M=lane%16
      a0.x = As[ar0][k0 + half * 2 + 0];
      a0.y = As[ar0][k0 + half * 2 + 1];
      a1.x = As[ar1][k0 + half * 2 + 0];
      a1.y = As[ar1][k0 + half * 2 + 1];
      // B 4x16 f32 layout (mirrored): N=lane%16, lanes 0-15 K={0,1}, 16-31 K={2,3}
      b0.x = Bs[bc0][k0 + half * 2 + 0];
      b0.y = Bs[bc0][k0 + half * 2 + 1];
      b1.x = Bs[bc1][k0 + half * 2 + 0];
      b1.y = Bs[bc1][k0 + half * 2 + 1];
      acc00 = __builtin_amdgcn_wmma_f32_16x16x4_f32(
          false, a0, false, b0, (short)0, acc00, false, false);
      acc01 = __builtin_amdgcn_wmma_f32_16x16x4_f32(
          false, a0, false, b1, (short)0, acc01, false, false);
      acc10 = __builtin_amdgcn_wmma_f32_16x16x4_f32(
          false, a1, false, b0, (short)0, acc10, false, false);
      acc11 = __builtin_amdgcn_wmma_f32_16x16x4_f32(
          false, a1, false, b1, (short)0, acc11, false, false);
    }
    __syncthreads();
  }

  // C/D layout: VGPR r -> M = r + 8*half, N = lane%16
  const int ncol0 = bn0 + wn * 32 + lr;
  const int ncol1 = ncol0 + 16;
  const float bias0 = bih[ncol0] + bhh[ncol0];
  const float bias1 = bih[ncol1] + bhh[ncol1];
#pragma unroll
  for (int r = 0; r < 8; ++r) {
    int m0 = bm0 + wm * 32 + half * 8 + r;
    int m1 = m0 + 16;
    gates[(size_t)m0 * kGates + ncol0] = acc00[r] + bias0;
    gates[(size_t)m0 * kGates + ncol1] = acc01[r] + bias1;
    gates[(size_t)m1 * kGates + ncol0] = acc10[r] + bias0;
    gates[(size_t)m1 * kGates + ncol1] = acc11[r] + bias1;
  }
}

// ---------------------------------------------------------------------------
// pointwise LSTM cell update with length masking (PyTorch gate order i,f,g,o)
// ---------------------------------------------------------------------------
__global__ void lstm_pointwise(const float* __restrict__ gates,
                               float* __restrict__ h,
                               float* __restrict__ c,
                               const int* __restrict__ len,
                               int t) {
  int g = blockIdx.x * blockDim.x + threadIdx.x;   // T*H threads
  int m  = g >> 8;
  int hh = g & 255;
  const float* gm = gates + (size_t)m * kGates;
  float gi = gm[hh];
  float gf = gm[kH + hh];
  float gg = gm[2 * kH + hh];
  float go = gm[3 * kH + hh];
  float si = 1.0f / (1.0f + expf(-gi));
  float sf = 1.0f / (1.0f + expf(-gf));
  float so = 1.0f / (1.0f + expf(-go));
  float cn = sf * c[g] + si * tanhf(gg);
  float hn = so * tanhf(cn);
  if (t < len[m]) {
    c[g] = cn;
    h[g] = hn;
  }
}

__global__ void seg_sum(const float* __restrict__ h,
                        const int* __restrict__ graph,
                        float* __restrict__ gbuf) {
  int g = blockIdx.x * blockDim.x + threadIdx.x;   // T*H threads
  int m  = g >> 8;
  int hh = g & 255;
  atomicAdd(&gbuf[(size_t)graph[m] * kH + hh], h[g]);
}

// score[b,o] += dot(gbuf[b,:], linW[o,:]) + linb[o]
__global__ void score_accum(const float* __restrict__ gbuf,
                            const float* __restrict__ linW,  // [OUT, H]
                            const float* __restrict__ linb,  // [OUT]
                            float* __restrict__ score) {
  int g = blockIdx.x * blockDim.x + threadIdx.x;   // B*OUT threads
  if (g >= kB * kOUT) return;
  int b = g >> 7;
  int o = g & 127;
  const float* gv = gbuf + (size_t)b * kH;
  const float* wv = linW + (size_t)o * kH;
  float s = 0.0f;
#pragma unroll 8
  for (int k = 0; k < kH; ++k) s += gv[k] * wv[k];
  score[g] += s + linb[o];
}

// ---------------------------------------------------------------------------
extern "C" void kernel_launch(void* const* d_in, const int* in_sizes, int n_in,
                              void* d_out, int out_size, void* d_ws, size_t ws_size,
                              hipStream_t stream) {
  (void)in_sizes; (void)n_in; (void)out_size; (void)ws_size;

  const float* emb0    = (const float*)d_in[0];
  const float* emb1    = (const float*)d_in[1];
  const float* bag_emb = (const float*)d_in[2];
  const float* pos_W   = (const float*)d_in[3];
  const float* pos_b   = (const float*)d_in[4];
  const float* W_ih_a  = (const float*)d_in[5];
  const float* W_hh_a  = (const float*)d_in[6];
  const float* b_ih_a  = (const float*)d_in[7];
  const float* b_hh_a  = (const float*)d_in[8];
  const float* W_ih_p  = (const float*)d_in[9];
  const float* W_hh_p  = (const float*)d_in[10];
  const float* b_ih_p  = (const float*)d_in[11];
  const float* b_hh_p  = (const float*)d_in[12];
  const float* lin_W_a = (const float*)d_in[13];
  const float* lin_b_a = (const float*)d_in[14];
  const float* lin_W_p = (const float*)d_in[15];
  const float* lin_b_p = (const float*)d_in[16];
  const float* x_pos   = (const float*)d_in[17];
  const int*   x_cat   = (const int*)d_in[18];
  // d_in[19] = row (arange(N)) -> identity gather, unused
  const int*   col        = (const int*)d_in[20];
  const int*   x_bag      = (const int*)d_in[21];
  const int*   trunk_idx  = (const int*)d_in[22];
  const int*   trunk_len  = (const int*)d_in[23];
  const int*   trunk_grph = (const int*)d_in[24];

  float* score = (float*)d_out;

  // workspace layout
  float* ws    = (float*)d_ws;
  float* xattr = ws;                                   // NB*H  (64 MB)
  float* xposb = xattr + (size_t)kNB * kH;             // NB*H  (64 MB)
  float* gates = xposb + (size_t)kNB * kH;             // T*4H  (8 MB)
  float* hbuf  = gates + (size_t)kT * kGates;          // T*H   (2 MB)
  float* cbuf  = hbuf + (size_t)kT * kH;               // T*H   (2 MB)
  float* gbuf  = cbuf + (size_t)kT * kH;               // B*H

  // 1) init outputs + bag matrices
  zero_f<<<(kB * kOUT + 255) / 256, 256, 0, stream>>>(score, kB * kOUT);
  init_bags<<<(kNB * kH) / 256, 256, 0, stream>>>(bag_emb, x_bag, xattr, xposb);
  node_scatter<<<((size_t)kN * kH) / 256, 256, 0, stream>>>(
      emb0, emb1, pos_W, pos_b, x_pos, x_cat, col, xattr, xposb);

  // 2) 4 levels x {attr, pos} LSTMs
  for (int l = 0; l < kLV; ++l) {
    const int* idx_l   = trunk_idx + (size_t)l * kT * kL;
    const int* len_l   = trunk_len + (size_t)l * kT;
    const int* graph_l = trunk_grph + (size_t)l * kT;

    for (int p = 0; p < 2; ++p) {
      const float* src  = (p == 0) ? xattr : xposb;
      const float* Wih  = ((p == 0) ? W_ih_a : W_ih_p) + (size_t)l * kGates * kH;
      const float* Whh  = ((p == 0) ? W_hh_a : W_hh_p) + (size_t)l * kGates * kH;
      const float* bih  = ((p == 0) ? b_ih_a : b_ih_p) + (size_t)l * kGates;
      const float* bhh  = ((p == 0) ? b_hh_a : b_hh_p) + (size_t)l * kGates;
      const float* linW = ((p == 0) ? lin_W_a : lin_W_p) + (size_t)l * kOUT * kH;
      const float* linb = ((p == 0) ? lin_b_a : lin_b_p) + (size_t)l * kOUT;

      zero_f<<<(kT * kH + 255) / 256, 256, 0, stream>>>(hbuf, kT * kH);
      zero_f<<<(kT * kH + 255) / 256, 256, 0, stream>>>(cbuf, kT * kH);

      dim3 ggrid(kT / BM, kGates / BN);
      for (int t = 0; t < kL; ++t) {
        lstm_gates_gemm<<<ggrid, 256, 0, stream>>>(
            src, idx_l, hbuf, Wih, Whh, bih, bhh, gates, t);
        lstm_pointwise<<<(kT * kH + 255) / 256, 256, 0, stream>>>(
            gates, hbuf, cbuf, len_l, t);
      }

      zero_f<<<(kB * kH + 255) / 256, 256, 0, stream>>>(gbuf, kB * kH);
      seg_sum<<<(kT * kH + 255) / 256, 256, 0, stream>>>(hbuf, graph_l, gbuf);
      score_accum<<<(kB * kOUT + 255) / 256, 256, 0, stream>>>(
          gbuf, linW, linb, score);
    }
  }
}